// StyleTransferNetwork_71227737637495
// MI455X (gfx1250) — compile-verified
//
#include <hip/hip_runtime.h>
#include <math.h>

typedef __attribute__((ext_vector_type(2))) float v2f;
typedef __attribute__((ext_vector_type(8))) float v8f;

// Problem shape (fixed by reference)
constexpr int B  = 16;
constexpr int C  = 512;
constexpr int HW = 64 * 64;          // 4096 pixels per (b,c)
constexpr float EPS = 1e-5f;

// Reduce-pass pixel chunking: 8 chunks x 512 px -> 8*32*16 = 4096 waves,
// enough concurrency to keep HBM busy while each wave still streams
// 16 channels x 512 px x 2 tensors (64 KB) through float4 loads.
constexpr int RCHUNKS = 8;
constexpr int RCHUNK_PX = HW / RCHUNKS;   // 512

// Workspace layout (floats)
constexpr int OFF_SXC  = 0;              // sum m*x      (content)  [B*C]
constexpr int OFF_SXXC = B * C;          // sum m*x^2    (content)  [B*C]
constexpr int OFF_SXS  = 2 * B * C;      // sum m*x      (style)    [B*C]
constexpr int OFF_SXXS = 3 * B * C;      // sum m*x^2    (style)    [B*C]
constexpr int OFF_CNTC = 4 * B * C;      // mask_c count per b      [B]
constexpr int OFF_CNTS = 4 * B * C + B;  // mask_s count per b      [B]
constexpr int WS_FLOATS = 4 * B * C + 2 * B;

__global__ void zero_ws_kernel(float* ws, int n) {
    int i = blockIdx.x * blockDim.x + threadIdx.x;
    if (i < n) ws[i] = 0.0f;
}

// Per-batch mask pixel counts (shared across all 512 channels).
__global__ __launch_bounds__(256)
void count_kernel(const int* __restrict__ mask_c, const int* __restrict__ mask_s,
                  float* __restrict__ ws) {
    const int b = blockIdx.x;
    __shared__ int sc[256];
    __shared__ int ss[256];
    const int t = threadIdx.x;
    int ac = 0, as = 0;
    for (int p = t; p < HW; p += 256) {
        ac += mask_c[b * HW + p];
        as += mask_s[b * HW + p];
    }
    sc[t] = ac; ss[t] = as;
    __syncthreads();
    for (int s = 128; s > 0; s >>= 1) {
        if (t < s) { sc[t] += sc[t + s]; ss[t] += ss[t + s]; }
        __syncthreads();
    }
    if (t == 0) {
        ws[OFF_CNTC + b] = (float)sc[0];
        ws[OFF_CNTS + b] = (float)ss[0];
    }
}

// One wave per (pixel-chunk, channel-group-of-16, batch).
// Lane L: channel row m = L&15, pixel half h = L>>4.
// Cross-lane channel sums are produced by V_WMMA_F32_16X16X4_F32 with B = ones:
// D[m,n] = sum_k A[m,k] -> row sums replicated across N.
__global__ __launch_bounds__(32)
void reduce_kernel(const float* __restrict__ content, const float* __restrict__ style,
                   const int* __restrict__ mask_c, const int* __restrict__ mask_s,
                   float* __restrict__ ws) {
    const int chunk = blockIdx.x;   // 0..RCHUNKS-1 (RCHUNK_PX pixels each)
    const int cg    = blockIdx.y;   // 0..31  (channel group of 16)
    const int b     = blockIdx.z;   // 0..15
    const int lane  = threadIdx.x;  // 0..31 (wave32)
    const int m     = lane & 15;
    const int h     = lane >> 4;
    const int c     = cg * 16 + m;
    const int p0    = chunk * RCHUNK_PX;

    const float* xc  = content + (size_t)(b * C + c) * HW;
    const float* xs  = style   + (size_t)(b * C + c) * HW;
    const int*   mcb = mask_c + b * HW;
    const int*   msb = mask_s + b * HW;

    float sxc = 0.0f, sxxc = 0.0f, sxs = 0.0f, sxxs = 0.0f;

    // Each lane-half covers RCHUNK_PX/2 pixels of the chunk via float4 loads.
#pragma unroll 4
    for (int p = p0 + h * 4; p < p0 + RCHUNK_PX; p += 8) {
        const float4 vc = *reinterpret_cast<const float4*>(xc + p);
        const float4 vs = *reinterpret_cast<const float4*>(xs + p);
        const int4   mc = *reinterpret_cast<const int4*>(mcb + p);
        const int4   ms = *reinterpret_cast<const int4*>(msb + p);
        float f;
        f = (float)mc.x; sxc += f * vc.x; sxxc += f * vc.x * vc.x;
        f = (float)mc.y; sxc += f * vc.y; sxxc += f * vc.y * vc.y;
        f = (float)mc.z; sxc += f * vc.z; sxxc += f * vc.z * vc.z;
        f = (float)mc.w; sxc += f * vc.w; sxxc += f * vc.w * vc.w;
        f = (float)ms.x; sxs += f * vs.x; sxxs += f * vs.x * vs.x;
        f = (float)ms.y; sxs += f * vs.y; sxxs += f * vs.y * vs.y;
        f = (float)ms.z; sxs += f * vs.z; sxxs += f * vs.z * vs.z;
        f = (float)ms.w; sxs += f * vs.w; sxxs += f * vs.w * vs.w;
    }

    // A is 16x4 f32: lanes 0-15 supply K=0 (vgpr0)/K=1 (vgpr1); lanes 16-31 K=2/K=3.
    // Put each lane's partial in its first A slot; second slot = 0.
    const v2f ones = {1.0f, 1.0f};
    const v8f zc   = {};
    v2f a;

    a = (v2f){sxc, 0.0f};
    v8f dxc  = __builtin_amdgcn_wmma_f32_16x16x4_f32(false, a, false, ones, (short)0, zc, false, false);
    a = (v2f){sxxc, 0.0f};
    v8f dxxc = __builtin_amdgcn_wmma_f32_16x16x4_f32(false, a, false, ones, (short)0, zc, false, false);
    a = (v2f){sxs, 0.0f};
    v8f dxs  = __builtin_amdgcn_wmma_f32_16x16x4_f32(false, a, false, ones, (short)0, zc, false, false);
    a = (v2f){sxxs, 0.0f};
    v8f dxxs = __builtin_amdgcn_wmma_f32_16x16x4_f32(false, a, false, ones, (short)0, zc, false, false);

    // D layout: vgpr e, lanes 0-15 -> row M=e; lanes 16-31 -> row M=e+8.
    // Lane 0 flushes channels cg*16+0..7, lane 16 flushes cg*16+8..15.
    if (m == 0) {
        const int cbase = b * C + cg * 16 + (h ? 8 : 0);
#pragma unroll
        for (int e = 0; e < 8; ++e) {
            unsafeAtomicAdd(&ws[OFF_SXC  + cbase + e], dxc[e]);
            unsafeAtomicAdd(&ws[OFF_SXXC + cbase + e], dxxc[e]);
            unsafeAtomicAdd(&ws[OFF_SXS  + cbase + e], dxs[e]);
            unsafeAtomicAdd(&ws[OFF_SXXS + cbase + e], dxxs[e]);
        }
    }
}

// Finalize stats (uniform per block -> scalar loads) + apply masked affine.
__global__ __launch_bounds__(256)
void apply_kernel(const float* __restrict__ content,
                  const int* __restrict__ mask_c,
                  const float* __restrict__ ws,
                  float* __restrict__ out) {
    const int b  = blockIdx.z;
    const int c  = blockIdx.y;
    const int p  = blockIdx.x * 1024 + threadIdx.x * 4;
    const int bc = b * C + c;

    const float cnc = ws[OFF_CNTC + b];
    const float cns = ws[OFF_CNTS + b];
    const float mean_c = ws[OFF_SXC + bc] / cnc;
    const float var_c  = (ws[OFF_SXXC + bc] - cnc * mean_c * mean_c) / (cnc - 1.0f);
    const float mean_s = ws[OFF_SXS + bc] / cns;
    const float var_s  = (ws[OFF_SXXS + bc] - cns * mean_s * mean_s) / (cns - 1.0f);
    const float std_c  = sqrtf(var_c + EPS);
    const float std_s  = sqrtf(var_s + EPS);
    const float scale  = std_s / std_c;
    const float shift  = mean_s - mean_c * scale;

    const size_t base = (size_t)bc * HW + p;
    const float4 x  = *reinterpret_cast<const float4*>(content + base);
    const int4   mm = *reinterpret_cast<const int4*>(mask_c + b * HW + p);

    float4 o;
    o.x = mm.x ? fmaf(x.x, scale, shift) : x.x;
    o.y = mm.y ? fmaf(x.y, scale, shift) : x.y;
    o.z = mm.z ? fmaf(x.z, scale, shift) : x.z;
    o.w = mm.w ? fmaf(x.w, scale, shift) : x.w;
    *reinterpret_cast<float4*>(out + base) = o;
}

extern "C" void kernel_launch(void* const* d_in, const int* in_sizes, int n_in,
                              void* d_out, int out_size, void* d_ws, size_t ws_size,
                              hipStream_t stream) {
    const float* content = (const float*)d_in[0];
    const float* style   = (const float*)d_in[1];
    const int*   mask_c  = (const int*)d_in[2];
    const int*   mask_s  = (const int*)d_in[3];
    float* out = (float*)d_out;
    float* ws  = (float*)d_ws;

    zero_ws_kernel<<<(WS_FLOATS + 255) / 256, 256, 0, stream>>>(ws, WS_FLOATS);
    count_kernel<<<B, 256, 0, stream>>>(mask_c, mask_s, ws);
    reduce_kernel<<<dim3(RCHUNKS, C / 16, B), 32, 0, stream>>>(content, style, mask_c, mask_s, ws);
    apply_kernel<<<dim3(HW / 1024, C, B), 256, 0, stream>>>(content, mask_c, ws, out);
}